// ResGCNEmbed_16458314678480
// MI455X (gfx1250) — compile-verified
//
#include <hip/hip_runtime.h>
#include <hip/hip_bf16.h>

#define H 128
#define H2 256
#define EPS_MSG 1e-7f
#define EPS_SM 1e-16f
#define LN_EPS 1e-5f

typedef float v2f __attribute__((ext_vector_type(2)));
typedef float v8f __attribute__((ext_vector_type(8)));

// -------------------- utility fills --------------------
__global__ __launch_bounds__(256) void fill_kernel(float* __restrict__ p, float v, long n) {
  long i = (long)blockIdx.x * blockDim.x + threadIdx.x;
  if (i < n) p[i] = v;
}

// -------------------- node encoder: x[N,8] @ W[8,128] + b --------------------
__global__ __launch_bounds__(128) void encoder_kernel(const float* __restrict__ x,
    const float* __restrict__ W, const float* __restrict__ b,
    float* __restrict__ out, int N) {
  long idx = (long)blockIdx.x * blockDim.x + threadIdx.x;
  if (idx >= (long)N * H) return;
  int n = (int)(idx >> 7);
  int c = (int)idx & (H - 1);
  const float* xr = x + (long)n * 8;
  float acc = b[c];
#pragma unroll
  for (int k = 0; k < 8; ++k) acc = fmaf(xr[k], W[k * H + c], acc);
  out[idx] = acc;
}

// -------------------- LayerNorm + ReLU, one wave32 per row --------------------
template<int W_>
__global__ __launch_bounds__(256) void ln_relu_kernel(const float* __restrict__ in,
    const float* __restrict__ g, const float* __restrict__ b,
    float* __restrict__ out, int N) {
  int wave = (int)(((long)blockIdx.x * blockDim.x + threadIdx.x) >> 5);
  int lane = threadIdx.x & 31;
  if (wave >= N) return;
  const int PER = W_ / 32;
  const float* row = in + (long)wave * W_;
  float v[PER];
  float s = 0.f;
#pragma unroll
  for (int i = 0; i < PER; ++i) { v[i] = row[lane + 32 * i]; s += v[i]; }
#pragma unroll
  for (int off = 16; off; off >>= 1) s += __shfl_xor(s, off, 32);
  float mu = s * (1.0f / (float)W_);
  float q = 0.f;
#pragma unroll
  for (int i = 0; i < PER; ++i) { float d = v[i] - mu; q = fmaf(d, d, q); }
#pragma unroll
  for (int off = 16; off; off >>= 1) q += __shfl_xor(q, off, 32);
  float rs = rsqrtf(q * (1.0f / (float)W_) + LN_EPS);
  float* orow = out + (long)wave * W_;
#pragma unroll
  for (int i = 0; i < PER; ++i) {
    float y = (v[i] - mu) * rs * g[lane + 32 * i] + b[lane + 32 * i];
    orow[lane + 32 * i] = fmaxf(y, 0.f);
  }
}

// -------------------- edge pass 1: segment max (float bits, all values > 0) ---
__global__ __launch_bounds__(256) void edge_max_kernel(const float* __restrict__ h,
    const int* __restrict__ src, const int* __restrict__ dst,
    const float* __restrict__ t, int layer, unsigned* __restrict__ mx, int E) {
  long tid = (long)blockIdx.x * blockDim.x + threadIdx.x;
  long e = tid >> 5;
  if (e >= E) return;
  int f = ((int)tid & 31) * 4;
  int s = src[e], d = dst[e];
  float tt = t[layer];
  const float4 hv = *(const float4*)(h + (long)s * H + f);
  unsigned* mp = mx + (long)d * H + f;
  float m0 = fmaxf(hv.x, 0.f) + EPS_MSG;
  float m1 = fmaxf(hv.y, 0.f) + EPS_MSG;
  float m2 = fmaxf(hv.z, 0.f) + EPS_MSG;
  float m3 = fmaxf(hv.w, 0.f) + EPS_MSG;
  atomicMax(mp + 0, __float_as_uint(m0 * tt));
  atomicMax(mp + 1, __float_as_uint(m1 * tt));
  atomicMax(mp + 2, __float_as_uint(m2 * tt));
  atomicMax(mp + 3, __float_as_uint(m3 * tt));
}

// -------------------- edge pass 2: exp + segment sums ------------------------
__global__ __launch_bounds__(256) void edge_sum_kernel(const float* __restrict__ h,
    const int* __restrict__ src, const int* __restrict__ dst,
    const float* __restrict__ t, int layer, const unsigned* __restrict__ mx,
    float* __restrict__ den, float* __restrict__ msum, int E) {
  long tid = (long)blockIdx.x * blockDim.x + threadIdx.x;
  long e = tid >> 5;
  if (e >= E) return;
  int f = ((int)tid & 31) * 4;
  int s = src[e], d = dst[e];
  float tt = t[layer];
  const float4 hv = *(const float4*)(h + (long)s * H + f);
  long o = (long)d * H + f;
  const float* mxf = (const float*)mx;
  float m0 = fmaxf(hv.x, 0.f) + EPS_MSG;
  float m1 = fmaxf(hv.y, 0.f) + EPS_MSG;
  float m2 = fmaxf(hv.z, 0.f) + EPS_MSG;
  float m3 = fmaxf(hv.w, 0.f) + EPS_MSG;
  float e0 = __expf(m0 * tt - mxf[o + 0]);
  float e1 = __expf(m1 * tt - mxf[o + 1]);
  float e2 = __expf(m2 * tt - mxf[o + 2]);
  float e3 = __expf(m3 * tt - mxf[o + 3]);
  atomicAdd(&den[o + 0], e0);
  atomicAdd(&den[o + 1], e1);
  atomicAdd(&den[o + 2], e2);
  atomicAdd(&den[o + 3], e3);
  atomicAdd(&msum[o + 0], m0 * e0);
  atomicAdd(&msum[o + 1], m1 * e1);
  atomicAdd(&msum[o + 2], m2 * e2);
  atomicAdd(&msum[o + 3], m3 * e3);
}

// -------------------- aggr/(den) + h -> h ------------------------------------
__global__ __launch_bounds__(256) void combine_kernel(const float* __restrict__ msum,
    const float* __restrict__ den, float* __restrict__ hb, int N) {
  long i = (long)blockIdx.x * blockDim.x + threadIdx.x;
  if (i >= (long)N * H) return;
  hb[i] = msum[i] / (den[i] + EPS_SM) + hb[i];
}

// -------------------- fp32 WMMA GEMM: D = A[M,K] @ B[K,Ncol] + bias (+ Cres) --
// Register-blocked: one wave32 computes a 64x16 slab of D (4 stacked 16x16
// tiles) via V_WMMA_F32_16X16X4_F32. Per k-step: 1 B fragment reused across
// 4 WMMAs, 4 A fragments (b64 loads) -> ~1.5 VMEM per WMMA instead of 3.
#define MB 4
__global__ __launch_bounds__(256) void gemm_wmma_f32(
    const float* __restrict__ A, const float* __restrict__ B,
    const float* __restrict__ bias, const float* __restrict__ Cres,
    float* __restrict__ D, int M, int K, int Ncol) {
  int wave = (int)(((long)blockIdx.x * blockDim.x + threadIdx.x) >> 5);
  int lane = threadIdx.x & 31;
  int tilesN = Ncol >> 4;
  int tilesM = (M + 16 * MB - 1) / (16 * MB);
  if (wave >= tilesM * tilesN) return;           // wave-uniform: EXEC stays all-1s
  int tm = wave / tilesN;
  int tn = wave - tm * tilesN;
  int row0 = tm * (16 * MB), col0 = tn << 4;
  int half = lane >> 4;                          // 0: K pair {0,1}, 1: K pair {2,3}
  int l = lane & 15;
  int kb = half * 2;

  const float* Ap[MB];
#pragma unroll
  for (int mi = 0; mi < MB; ++mi) {
    int arow = min(row0 + mi * 16 + l, M - 1);
    Ap[mi] = A + (long)arow * K;                 // A frag: lane holds row, K=kb,kb+1
  }
  const float* Bp = B + col0 + l;                // B frag: lane holds col, K=kb,kb+1
  int col = col0 + l;
  float bb = bias[col];

  v8f c[MB];
#pragma unroll
  for (int mi = 0; mi < MB; ++mi) {
#pragma unroll
    for (int r = 0; r < 8; ++r) {
      int rr = row0 + mi * 16 + r + 8 * half;
      float cv = bb;
      if (Cres != nullptr && rr < M) cv += Cres[(long)rr * Ncol + col];  // fused residual
      c[mi][r] = cv;
    }
  }

  for (int k = 0; k < K; k += 4) {
    v2f b;
    b.x = Bp[(long)(k + kb) * Ncol];
    b.y = Bp[(long)(k + kb + 1) * Ncol];
#pragma unroll
    for (int mi = 0; mi < MB; ++mi) {
      v2f a;
      a.x = Ap[mi][k + kb];
      a.y = Ap[mi][k + kb + 1];
      c[mi] = __builtin_amdgcn_wmma_f32_16x16x4_f32(false, a, false, b, (short)0, c[mi],
                                                    false, false);
    }
  }

#pragma unroll
  for (int mi = 0; mi < MB; ++mi) {
#pragma unroll
    for (int r = 0; r < 8; ++r) {
      int rr = row0 + mi * 16 + r + 8 * half;
      if (rr < M) D[(long)rr * Ncol + col] = c[mi][r];
    }
  }
}

// -------------------- global add pool ----------------------------------------
__global__ __launch_bounds__(256) void pool_kernel(const float* __restrict__ x,
    const int* __restrict__ bid, float* __restrict__ out, int N) {
  long i = (long)blockIdx.x * blockDim.x + threadIdx.x;
  if (i >= (long)N * H) return;
  int n = (int)(i >> 7);
  int f = (int)i & (H - 1);
  atomicAdd(&out[(long)bid[n] * H + f], x[i]);
}

// =============================================================================
extern "C" void kernel_launch(void* const* d_in, const int* in_sizes, int n_in,
                              void* d_out, int out_size, void* d_ws, size_t ws_size,
                              hipStream_t stream) {
  const float* x_in   = (const float*)d_in[0];
  const int*   eidx   = (const int*)d_in[1];
  const int*   batch  = (const int*)d_in[2];
  const float* node_W = (const float*)d_in[3];
  const float* node_b = (const float*)d_in[4];
  const float* ln_g   = (const float*)d_in[5];
  const float* ln_b   = (const float*)d_in[6];
  const float* t_arr  = (const float*)d_in[7];
  const float* W1     = (const float*)d_in[8];
  const float* b1     = (const float*)d_in[9];
  const float* mln_g  = (const float*)d_in[10];
  const float* mln_b  = (const float*)d_in[11];
  const float* W2     = (const float*)d_in[12];
  const float* b2     = (const float*)d_in[13];

  const int N = in_sizes[0] / 8;   // 50000
  const int E = in_sizes[1] / 2;   // 800000
  const int L = in_sizes[7];       // 6
  const int* src = eidx;
  const int* dst = eidx + E;

  const long NH = (long)N * H;
  float* X    = (float*)d_ws;      // [N,H]
  float* Hb   = X + NH;            // [N,H]
  float* MX   = Hb + NH;           // [N,H] (uint-bit max accumulator)
  float* DEN  = MX + NH;           // [N,H]
  float* MSUM = DEN + NH;          // [N,H]
  float* U    = MSUM + NH;         // [N,2H]

  // node encoder
  encoder_kernel<<<(int)((NH + 127) / 128), 128, 0, stream>>>(x_in, node_W, node_b, X, N);

  const int lnBlocks   = (int)(((long)N * 32 + 255) / 256);
  const int edgeBlocks = (int)(((long)E * 32 + 255) / 256);

  for (int l = 0; l < L; ++l) {
    // h = relu(LN(x))
    ln_relu_kernel<H><<<lnBlocks, 256, 0, stream>>>(X, ln_g + (long)l * H, ln_b + (long)l * H, Hb, N);
    // zero MX/DEN/MSUM in one contiguous fill (0 bits == 0.0f, valid max-identity for positives)
    fill_kernel<<<(int)((3 * NH + 255) / 256), 256, 0, stream>>>(MX, 0.f, 3 * NH);
    // softmax aggregation over dst segments
    edge_max_kernel<<<edgeBlocks, 256, 0, stream>>>(Hb, src, dst, t_arr, l, (unsigned*)MX, E);
    edge_sum_kernel<<<edgeBlocks, 256, 0, stream>>>(Hb, src, dst, t_arr, l,
                                                    (const unsigned*)MX, DEN, MSUM, E);
    // h = aggr + h
    combine_kernel<<<(int)((NH + 255) / 256), 256, 0, stream>>>(MSUM, DEN, Hb, N);
    // U = h @ W1 + b1
    {
      int waves = ((N + 16 * MB - 1) / (16 * MB)) * (H2 / 16);
      gemm_wmma_f32<<<(int)(((long)waves * 32 + 255) / 256), 256, 0, stream>>>(
          Hb, W1 + (long)l * H * H2, b1 + (long)l * H2, nullptr, U, N, H, H2);
    }
    // U = relu(LN(U))
    ln_relu_kernel<H2><<<lnBlocks, 256, 0, stream>>>(U, mln_g + (long)l * H2, mln_b + (long)l * H2, U, N);
    // X = X + U @ W2 + b2   (residual fused into WMMA C)
    {
      int waves = ((N + 16 * MB - 1) / (16 * MB)) * (H / 16);
      gemm_wmma_f32<<<(int)(((long)waves * 32 + 255) / 256), 256, 0, stream>>>(
          U, W2 + (long)l * H2 * H, b2 + (long)l * H, X, X, N, H2, H);
    }
  }

  // global add pool -> d_out [64, H]
  fill_kernel<<<(out_size + 255) / 256, 256, 0, stream>>>((float*)d_out, 0.f, out_size);
  pool_kernel<<<(int)((NH + 255) / 256), 256, 0, stream>>>(X, batch, (float*)d_out, N);
}